// CapsuleNetwork_64355789963357
// MI455X (gfx1250) — compile-verified
//
#include <hip/hip_runtime.h>
#include <hip/hip_bf16.h>

typedef __attribute__((ext_vector_type(2))) float v2f;
typedef __attribute__((ext_vector_type(8))) float v8f;

#define EPS 1e-7f

__device__ __forceinline__ v8f wmma4(v2f a, v2f b, v8f c) {
  // D = A(16x4) * B(4x16) + C(16x16), all fp32
  return __builtin_amdgcn_wmma_f32_16x16x4_f32(false, a, false, b, (short)0, c,
                                               false, false);
}

__device__ __forceinline__ float act_f(float v, int act) {
  if (act == 1) return v > 0.f ? v : 0.f;
  if (act == 2) return 1.f / (1.f + __expf(-v));
  return v;
}

// ---------------------------------------------------------------------------
// conv1: (B,28,28,1) -> relu -> (B,20,20,256).  9x9 VALID, stride 1.
// One block per (b, oy); thread t = output channel; 20 ox accumulators.
// Each weight read exactly once per block (coalesced over channels).
// ---------------------------------------------------------------------------
__global__ void conv1_kernel(const float* __restrict__ img,
                             const float* __restrict__ Wt,   // (9,9,1,256)
                             const float* __restrict__ bias,
                             float* __restrict__ Y) {
  int b = blockIdx.y, oy = blockIdx.x, t = threadIdx.x;
  __shared__ float srow[9 * 28];
  const float* ib = img + b * 784;
  for (int i = t; i < 9 * 28; i += 256)
    srow[i] = ib[(oy + i / 28) * 28 + (i % 28)];
  __syncthreads();

  float acc[20];
#pragma unroll
  for (int ox = 0; ox < 20; ++ox) acc[ox] = 0.f;

  for (int ky = 0; ky < 9; ++ky)
#pragma unroll
    for (int kx = 0; kx < 9; ++kx) {
      float wv = Wt[(ky * 9 + kx) * 256 + t];
#pragma unroll
      for (int ox = 0; ox < 20; ++ox)
        acc[ox] = fmaf(srow[ky * 28 + ox + kx], wv, acc[ox]);
    }
  float bv = bias[t];
#pragma unroll
  for (int ox = 0; ox < 20; ++ox) {
    float v = acc[ox] + bv;
    Y[((b * 20 + oy) * 20 + ox) * 256 + t] = v > 0.f ? v : 0.f;
  }
}

// ---------------------------------------------------------------------------
// conv2 as implicit GEMM with fp32 WMMA.
// M = 256*36 = 9216 (b,oy,ox), N = 256 (oc), K = 81*256 = 20736 (ky,kx,ic).
// Block: 256 threads (8 wave32), 64x64 C tile, BK=16 LDS staging.
// Each wave: 16 rows x 32 cols (two 16x16 v8f accumulators).
// ---------------------------------------------------------------------------
__global__ void conv2_wmma_kernel(const float* __restrict__ X,   // (256,20,20,256)
                                  const float* __restrict__ Wt,  // (9,9,256,256)
                                  const float* __restrict__ bias,
                                  float* __restrict__ Y) {       // (9216,256)
  __shared__ float As[64][17];  // +1 pad: avoid 4-way bank conflicts
  __shared__ float Bs[16][64];

  int t = threadIdx.x;
  int m0 = blockIdx.x * 64, n0 = blockIdx.y * 64;
  int wid = t >> 5, lane = t & 31;
  int wm = wid >> 1, wn = wid & 1;
  int ml = lane & 15, kb = (lane >> 4) << 1;

  // Precompute per-thread im2col row bases (stage-invariant).
  int abase[4], arow[4], akl[4], brow[4], bcol[4];
#pragma unroll
  for (int e = 0; e < 4; ++e) {
    int lin = t + e * 256;
    arow[e] = lin >> 4;
    akl[e] = lin & 15;
    int m = m0 + arow[e];
    int b = m / 36;
    int sp = m - b * 36;
    int oy = sp / 6, ox = sp - oy * 6;
    abase[e] = ((b * 20 + 2 * oy) * 20 + 2 * ox) * 256;
    brow[e] = lin >> 6;
    bcol[e] = lin & 63;
  }

  v8f c0 = {0.f, 0.f, 0.f, 0.f, 0.f, 0.f, 0.f, 0.f};
  v8f c1 = {0.f, 0.f, 0.f, 0.f, 0.f, 0.f, 0.f, 0.f};

  for (int k0 = 0; k0 < 20736; k0 += 16) {
    int kykx = k0 >> 8;          // one (ky,kx) per stage (256 % 16 == 0)
    int ic0 = k0 & 255;
    int ky = kykx / 9, kx = kykx - ky * 9;
    int off = (ky * 20 + kx) * 256 + ic0;
#pragma unroll
    for (int e = 0; e < 4; ++e)
      As[arow[e]][akl[e]] = X[abase[e] + off + akl[e]];
#pragma unroll
    for (int e = 0; e < 4; ++e)
      Bs[brow[e]][bcol[e]] = Wt[(k0 + brow[e]) * 256 + n0 + bcol[e]];
    __syncthreads();

#pragma unroll
    for (int kk = 0; kk < 4; ++kk) {
      v2f a;
      a.x = As[16 * wm + ml][4 * kk + kb];
      a.y = As[16 * wm + ml][4 * kk + kb + 1];
      int col0 = 32 * wn + ml;
      v2f b0;
      b0.x = Bs[4 * kk + kb][col0];
      b0.y = Bs[4 * kk + kb + 1][col0];
      c0 = wmma4(a, b0, c0);
      v2f b1;
      b1.x = Bs[4 * kk + kb][col0 + 16];
      b1.y = Bs[4 * kk + kb + 1][col0 + 16];
      c1 = wmma4(a, b1, c1);
    }
    __syncthreads();
  }

  int mb = m0 + 16 * wm + ((lane >> 4) << 3);
#pragma unroll
  for (int r = 0; r < 8; ++r) {
    int m = mb + r;
    int n = n0 + 32 * wn + ml;
    float v = c0[r] + bias[n];
    Y[m * 256 + n] = v > 0.f ? v : 0.f;
    n += 16;
    v = c1[r] + bias[n];
    Y[m * 256 + n] = v > 0.f ? v : 0.f;
  }
}

// ---------------------------------------------------------------------------
// squash over last dim of (B*1152, 8)
// ---------------------------------------------------------------------------
__global__ void squash8_kernel(const float* __restrict__ in,
                               float* __restrict__ out, int ncaps) {
  int i = blockIdx.x * blockDim.x + threadIdx.x;
  if (i >= ncaps) return;
  const float* s = in + i * 8;
  float v[8], sq = 0.f;
#pragma unroll
  for (int j = 0; j < 8; ++j) {
    v[j] = s[j];
    sq = fmaf(v[j], v[j], sq);
  }
  float scale = sq / ((1.f + sq) * sqrtf(sq + EPS));
#pragma unroll
  for (int j = 0; j < 8; ++j) out[i * 8 + j] = v[j] * scale;
}

// ---------------------------------------------------------------------------
// Fused capsule routing per batch element b (one block each, 256 threads,
// 160 active as (class c, dim j)).  u_hat recomputed in pass 2 to avoid
// materializing the 189 MB u_hat tensor.  Emits norm_v2, y_pred (as float),
// and masked flat (B,160) for the decoder.
// ---------------------------------------------------------------------------
__global__ void routing_kernel(const float* __restrict__ pc,   // (B,1152,8)
                               const float* __restrict__ W,    // (1152,10,16,8)
                               const int* __restrict__ target,
                               float* __restrict__ out_norm,   // (B,10)
                               float* __restrict__ out_pred,   // (B,)
                               float* __restrict__ masked) {   // (B,160)
  int b = blockIdx.x, t = threadIdx.x;
  __shared__ float pcsh[1152 * 8];
  __shared__ float shv[160];
  __shared__ float shu[160];
  __shared__ float shuv[10];
  __shared__ float shnorm[10];

  const float* pcb = pc + b * 9216;
  for (int i = t; i < 9216; i += 256) pcsh[i] = pcb[i];
  __syncthreads();

  bool active = t < 160;
  int c = t >> 4, j = t & 15;

  // pass 1: s1[c][j] = (1/10) * sum_n u_hat
  float s1 = 0.f;
  if (active) {
    for (int n = 0; n < 1152; ++n) {
      const float* wn = W + ((n * 10 + c) * 16 + j) * 8;
      const float* p = pcsh + n * 8;
      float u = 0.f;
#pragma unroll
      for (int i = 0; i < 8; ++i) u = fmaf(wn[i], p[i], u);
      s1 += u;
    }
    s1 *= 0.1f;
    shv[t] = s1;
  }
  __syncthreads();
  float v1own = 0.f;
  if (active) {
    float sq = 0.f;
    for (int jj = 0; jj < 16; ++jj) {
      float x = shv[c * 16 + jj];
      sq = fmaf(x, x, sq);
    }
    v1own = s1 * (sq / ((1.f + sq) * sqrtf(sq + EPS)));
  }
  __syncthreads();

  // pass 2: agreement -> softmax over classes -> s2
  float s2 = 0.f;
  for (int n = 0; n < 1152; ++n) {
    float u = 0.f;
    if (active) {
      const float* wn = W + ((n * 10 + c) * 16 + j) * 8;
      const float* p = pcsh + n * 8;
#pragma unroll
      for (int i = 0; i < 8; ++i) u = fmaf(wn[i], p[i], u);
      shu[t] = u * v1own;
    }
    __syncthreads();
    if (active && j == 0) {
      float uv = 0.f;
      for (int jj = 0; jj < 16; ++jj) uv += shu[c * 16 + jj];
      shuv[c] = uv;
    }
    __syncthreads();
    if (active) {
      float mx = shuv[0];
      for (int cc = 1; cc < 10; ++cc) mx = fmaxf(mx, shuv[cc]);
      float sum = 0.f;
      for (int cc = 0; cc < 10; ++cc) sum += __expf(shuv[cc] - mx);
      float c2 = __expf(shuv[c] - mx) / sum;
      s2 = fmaf(c2, u, s2);
    }
    __syncthreads();
  }

  // v2 = squash(s2), norm, argmax, mask
  if (active) shv[t] = s2;
  __syncthreads();
  if (active) {
    float sq = 0.f;
    for (int jj = 0; jj < 16; ++jj) {
      float x = shv[c * 16 + jj];
      sq = fmaf(x, x, sq);
    }
    float scale = sq / ((1.f + sq) * sqrtf(sq + EPS));
    float v2own = s2 * scale;
    if (j == 0) {
      float nv = sqrtf(sq * scale * scale + EPS);  // ||v2|| with eps
      shnorm[c] = nv;
      out_norm[b * 10 + c] = nv;
    }
    int tgt = target[b];
    masked[b * 160 + t] = (c == tgt) ? v2own : 0.f;
  }
  __syncthreads();
  if (t == 0) {
    int am = 0;
    float bm = shnorm[0];
    for (int cc = 1; cc < 10; ++cc)
      if (shnorm[cc] > bm) { bm = shnorm[cc]; am = cc; }
    out_pred[b] = (float)am;
  }
}

// ---------------------------------------------------------------------------
// Generic row-major GEMM + bias + activation with fp32 WMMA.
// D(MxN) = act(A(MxK) * B(KxN) + bias).  M%64==0, K%16==0; N guarded.
// ---------------------------------------------------------------------------
__global__ void gemm_wmma_kernel(const float* __restrict__ A,
                                 const float* __restrict__ Bm,
                                 const float* __restrict__ bias,
                                 float* __restrict__ D,
                                 int M, int N, int K, int act) {
  __shared__ float As[64][17];
  __shared__ float Bs[16][64];

  int t = threadIdx.x;
  int m0 = blockIdx.x * 64, n0 = blockIdx.y * 64;
  int wid = t >> 5, lane = t & 31;
  int wm = wid >> 1, wn = wid & 1;
  int ml = lane & 15, kb = (lane >> 4) << 1;

  v8f c0 = {0.f, 0.f, 0.f, 0.f, 0.f, 0.f, 0.f, 0.f};
  v8f c1 = {0.f, 0.f, 0.f, 0.f, 0.f, 0.f, 0.f, 0.f};

  for (int k0 = 0; k0 < K; k0 += 16) {
#pragma unroll
    for (int e = 0; e < 4; ++e) {
      int lin = t + e * 256;
      As[lin >> 4][lin & 15] = A[(m0 + (lin >> 4)) * K + k0 + (lin & 15)];
    }
#pragma unroll
    for (int e = 0; e < 4; ++e) {
      int lin = t + e * 256;
      int r = lin >> 6, col = lin & 63;
      int n = n0 + col;
      Bs[r][col] = (n < N) ? Bm[(k0 + r) * N + n] : 0.f;
    }
    __syncthreads();

#pragma unroll
    for (int kk = 0; kk < 4; ++kk) {
      v2f a;
      a.x = As[16 * wm + ml][4 * kk + kb];
      a.y = As[16 * wm + ml][4 * kk + kb + 1];
      int col0 = 32 * wn + ml;
      v2f b0;
      b0.x = Bs[4 * kk + kb][col0];
      b0.y = Bs[4 * kk + kb + 1][col0];
      c0 = wmma4(a, b0, c0);
      v2f b1;
      b1.x = Bs[4 * kk + kb][col0 + 16];
      b1.y = Bs[4 * kk + kb + 1][col0 + 16];
      c1 = wmma4(a, b1, c1);
    }
    __syncthreads();
  }

  int mb = m0 + 16 * wm + ((lane >> 4) << 3);
#pragma unroll
  for (int r = 0; r < 8; ++r) {
    int m = mb + r;
    int n = n0 + 32 * wn + ml;
    if (n < N) D[m * N + n] = act_f(c0[r] + bias[n], act);
    n += 16;
    if (n < N) D[m * N + n] = act_f(c1[r] + bias[n], act);
  }
}

// ---------------------------------------------------------------------------
extern "C" void kernel_launch(void* const* d_in, const int* in_sizes, int n_in,
                              void* d_out, int out_size, void* d_ws,
                              size_t ws_size, hipStream_t stream) {
  const float* image = (const float*)d_in[0];
  const int* target  = (const int*)d_in[1];
  const float* c1w   = (const float*)d_in[2];
  const float* c1b   = (const float*)d_in[3];
  const float* c2w   = (const float*)d_in[4];
  const float* c2b   = (const float*)d_in[5];
  const float* Wr    = (const float*)d_in[6];
  const float* d1w   = (const float*)d_in[7];
  const float* d1b   = (const float*)d_in[8];
  const float* d2w   = (const float*)d_in[9];
  const float* d2b   = (const float*)d_in[10];
  const float* d3w   = (const float*)d_in[11];
  const float* d3b   = (const float*)d_in[12];

  float* out = (float*)d_out;
  float* ws = (float*)d_ws;

  // Workspace layout (floats)
  float* h1    = ws;                    // 256*20*20*256 = 26,214,400
  float* h2    = h1 + 26214400;         // 9216*256      =  2,359,296
  float* pc    = h2 + 2359296;          //                  2,359,296
  float* mflat = pc + 2359296;          // 256*160       =     40,960
  float* dh1   = mflat + 40960;         // 256*512       =    131,072
  float* dh2   = dh1 + 131072;          // 256*1024      =    262,144

  float* out_norm = out;         // (256,1,10,1,1) = 2560
  float* out_pred = out + 2560;  // (256,)         =  256
  float* out_dec  = out + 2816;  // (256,784)      = 200704

  conv1_kernel<<<dim3(20, 256), 256, 0, stream>>>(image, c1w, c1b, h1);
  conv2_wmma_kernel<<<dim3(144, 4), 256, 0, stream>>>(h1, c2w, c2b, h2);
  squash8_kernel<<<1152, 256, 0, stream>>>(h2, pc, 294912);
  routing_kernel<<<256, 256, 0, stream>>>(pc, Wr, target, out_norm, out_pred,
                                          mflat);
  gemm_wmma_kernel<<<dim3(4, 8), 256, 0, stream>>>(mflat, d1w, d1b, dh1,
                                                   256, 512, 160, 1);
  gemm_wmma_kernel<<<dim3(4, 16), 256, 0, stream>>>(dh1, d2w, d2b, dh2,
                                                    256, 1024, 512, 1);
  gemm_wmma_kernel<<<dim3(4, 13), 256, 0, stream>>>(dh2, d3w, d3b, out_dec,
                                                    256, 784, 1024, 2);
}